// GridInterpolateRouter_77945066488039
// MI455X (gfx1250) — compile-verified
//
#include <hip/hip_runtime.h>
#include <hip/hip_bf16.h>

// ---------------- problem constants (from reference) ----------------
#define H_DIM   4096
#define NTOK    8192
#define NOUT    12          // M_ANCH*(D+1)
#define NPAD    16          // padded N for WMMA
#define EGRID   64          // 8*8 grid cells
#define KTOP    16
#define MANCH   4
#define CHUNKS  (H_DIM / 4) // 1024 K-chunks of 4 for wmma f32 16x16x4

// GEMM tiling
#define BM      128         // tokens per workgroup (8 waves x 16)
#define BK      64          // K per LDS tile
#define NTILE   (H_DIM / BK)
#define LDSROW  68          // dwords per LDS row (16B-aligned rows, conflict-free)

typedef float v2f __attribute__((ext_vector_type(2)));
typedef float v4f __attribute__((ext_vector_type(4)));
typedef float v8f __attribute__((ext_vector_type(8)));

// ---------------------------------------------------------------------------
// Kernel 0: pack W (12x4096, zero-padded to 16) into B-fragment layout.
// Entry g = chunk*32 + lane. lane L: N = L&15, K-half = 2*(L>>4).
// B[K][N] = W[N][K]  (logits = hidden @ W^T)
// ---------------------------------------------------------------------------
__global__ __launch_bounds__(256) void pack_B_kernel(const float* __restrict__ W,
                                                     float2* __restrict__ Bpack) {
    int g = blockIdx.x * 256 + threadIdx.x;
    if (g >= CHUNKS * 32) return;
    int c = g >> 5;
    int L = g & 31;
    int n = L & 15;
    int k = c * 4 + ((L >> 4) * 2);
    float2 v = make_float2(0.0f, 0.0f);
    if (n < NOUT) {
        v.x = W[n * H_DIM + k];
        v.y = W[n * H_DIM + k + 1];
    }
    Bpack[g] = v;
}

// ---------------------------------------------------------------------------
// Kernel 1: logits(8192x16) = hidden @ W^T + b via V_WMMA_F32_16X16X4_F32.
// 256 threads = 8 waves; each wave owns a 16-token x 16-N tile AND streams its
// own 16 rows via GLOBAL_LOAD_ASYNC_TO_LDS_B128 (double-buffered, ASYNCcnt).
// No cross-wave synchronization; last tile peeled so the hot loop is
// branch-free: issue next -> s_wait_asynccnt 8 -> 16x WMMA.
// ---------------------------------------------------------------------------
__global__ __launch_bounds__(256) void gemm_wmma_kernel(const float* __restrict__ hidden,
                                                        const float* __restrict__ Bpack,
                                                        const float* __restrict__ bias,
                                                        float* __restrict__ logits) {
    // [buffer][wave][16 rows * LDSROW]  -> 2*8*16*68*4 = 68 KB
    __shared__ __align__(16) float tile[2][8][16 * LDSROW];

    const int tid   = threadIdx.x;
    const int wave  = tid >> 5;
    const int lane  = tid & 31;
    const int rowBase = blockIdx.x * BM + wave * 16;   // this wave's 16 tokens
    const int laneM = lane & 15;
    const int koff  = (lane >> 4) * 2;   // K half-wave split (A and B frags)
    const int c4    = lane & 15;         // float4 column this lane fills
    const int rhalf = lane >> 4;         // row parity this lane fills

    // low 32 bits of a generic shared pointer == LDS offset (aperture rule)
    uint32_t ldsw[2];
    ldsw[0] = (uint32_t)(uintptr_t)&tile[0][wave][0];
    ldsw[1] = (uint32_t)(uintptr_t)&tile[1][wave][0];
    const uint64_t gwave = (uint64_t)(uintptr_t)(hidden + (size_t)rowBase * H_DIM);

    // one 16x64 tile fill for this wave: 8 async b128 loads per lane, each
    // instruction covers 2 rows x 256 contiguous bytes (coalesced, NT).
    auto issue_tile = [&](int b, int k0) {
        const uint32_t lbase = ldsw[b];
        #pragma unroll
        for (int i = 0; i < 8; ++i) {
            int r = 2 * i + rhalf;
            uint32_t loff = lbase + (uint32_t)(r * LDSROW + c4 * 4) * 4u;
            uint64_t ga   = gwave + ((uint64_t)r * H_DIM + (uint64_t)k0 + c4 * 4) * 4u;
            asm volatile("global_load_async_to_lds_b128 %0, %1, off th:TH_LOAD_NT"
                         :: "v"(loff), "v"(ga) : "memory");
        }
    };

    v8f acc = {0.f, 0.f, 0.f, 0.f, 0.f, 0.f, 0.f, 0.f};

    // 16 WMMAs over one resident tile
    auto compute_tile = [&](int b, int k0) {
        const float* tw = &tile[b][wave][0];
        #pragma unroll
        for (int kk = 0; kk < BK; kk += 4) {
            // A fragment: lane holds hidden[row = rowBase+laneM][k+koff .. +1]
            v2f a = *(const v2f*)&tw[laneM * LDSROW + kk + koff];
            // B fragment: L2-resident pack, one b64 per lane
            v2f b2 = *(const v2f*)&Bpack[((size_t)((k0 + kk) >> 2) * 32 + lane) * 2];
            acc = __builtin_amdgcn_wmma_f32_16x16x4_f32(
                /*neg_a=*/false, a, /*neg_b=*/false, b2,
                /*c_mod=*/(short)0, acc, /*reuse_a=*/false, /*reuse_b=*/false);
        }
    };

    issue_tile(0, 0);
    int k0 = 0;
    #pragma unroll 1
    for (int t = 0; t < NTILE - 1; ++t, k0 += BK) {
        const int cur = t & 1;
        issue_tile(cur ^ 1, k0 + BK);                       // prefetch next tile
        asm volatile("s_wait_asynccnt 0x8" ::: "memory");   // cur tile resident
        compute_tile(cur, k0);
    }
    asm volatile("s_wait_asynccnt 0x0" ::: "memory");       // last tile resident
    compute_tile((NTILE - 1) & 1, k0);

    // epilogue: C layout -> lane N = lane&15 ; VGPR v -> M = 8*(lane>>4)+v
    const int N = lane & 15;
    const int mBase = (lane >> 4) * 8;
    float bn = (N < NOUT) ? bias[N] : 0.0f;
    #pragma unroll
    for (int v = 0; v < 8; ++v) {
        int tok = rowBase + mBase + v;
        logits[tok * NPAD + N] = acc[v] + bn;
    }
}

// ---------------------------------------------------------------------------
// Kernel 2: per-token routing + top-16 of 64. One thread per token.
// 64-cell histogram in LDS (stride 65 -> bank-conflict free).
// ---------------------------------------------------------------------------
__global__ __launch_bounds__(128) void route_topk_kernel(const float* __restrict__ logits,
                                                         int* __restrict__ out_idx,
                                                         float* __restrict__ out_w) {
    __shared__ float probs[128 * 65];
    const int tid = threadIdx.x;
    const int tok = blockIdx.x * 128 + tid;
    float* pr = &probs[tid * 65];

    #pragma unroll
    for (int i = 0; i < EGRID; ++i) pr[i] = 0.0f;

    float l[NOUT];
    #pragma unroll
    for (int j = 0; j < NOUT; ++j) l[j] = logits[tok * NPAD + j];

    // anchor softmax over l[a*3+2], a=0..3
    float al[MANCH];
    #pragma unroll
    for (int a = 0; a < MANCH; ++a) al[a] = l[a * 3 + 2];
    float am = fmaxf(fmaxf(al[0], al[1]), fmaxf(al[2], al[3]));
    float ea[MANCH], es = 0.0f;
    #pragma unroll
    for (int a = 0; a < MANCH; ++a) { ea[a] = expf(al[a] - am); es += ea[a]; }
    float inv_es = 1.0f / es;

    const float EPSC = 1e-6f;
    #pragma unroll
    for (int a = 0; a < MANCH; ++a) {
        float pi = ea[a] * inv_es;
        float u0 = 1.0f / (1.0f + expf(-l[a * 3 + 0]));
        float u1 = 1.0f / (1.0f + expf(-l[a * 3 + 1]));
        u0 = fminf(fmaxf(u0, EPSC), 1.0f - EPSC);
        u1 = fminf(fmaxf(u1, EPSC), 1.0f - EPSC);
        float p0 = fmaxf(fminf(u0 * 7.0f, 7.0f - 1e-6f), 0.0f);
        float p1 = fmaxf(fminf(u1 * 7.0f, 7.0f - 1e-6f), 0.0f);
        int a0 = (int)floorf(p0); a0 = min(max(a0, 0), 6);
        int a1 = (int)floorf(p1); a1 = min(max(a1, 0), 6);
        float f0 = fminf(fmaxf(p0 - (float)a0, EPSC), 1.0f - EPSC);
        float f1 = fminf(fmaxf(p1 - (float)a1, EPSC), 1.0f - EPSC);
        float w00 = (1.0f - f0) * (1.0f - f1);
        float w10 = f0 * (1.0f - f1);
        float w01 = (1.0f - f0) * f1;
        float w11 = f0 * f1;
        float wsum = w00 + w10 + w01 + w11;
        float s = pi / (wsum + 1e-9f);
        int base = a0 + 8 * a1;          // strides = [1, 8]
        pr[base]     += w00 * s;
        pr[base + 1] += w10 * s;
        pr[base + 8] += w01 * s;
        pr[base + 9] += w11 * s;
    }

    // normalize
    float psum = 0.0f;
    #pragma unroll
    for (int i = 0; i < EGRID; ++i) { float v = fmaxf(pr[i], 0.0f); pr[i] = v; psum += v; }
    float inv = 1.0f / (psum + 1e-9f);
    #pragma unroll
    for (int i = 0; i < EGRID; ++i) pr[i] *= inv;

    // top-16 by repeated argmax (strict '>' keeps lowest index on ties, matching lax.top_k)
    for (int kk = 0; kk < KTOP; ++kk) {
        float best = -1.0f;
        int   bi   = -1;
        for (int i = 0; i < EGRID; ++i) {
            float v = pr[i];
            if (v > best) { best = v; bi = i; }
        }
        pr[bi] = -2.0f;
        out_idx[tok * KTOP + kk] = bi;
        out_w[tok * KTOP + kk]   = best;
    }
}

// ---------------------------------------------------------------------------
// Launch. ws layout: [Bpack: 256 KB][logits16: 512 KB]  (768 KB total)
// d_out: [top_idx int32 8192x16][top_w f32 8192x16]
// ---------------------------------------------------------------------------
extern "C" void kernel_launch(void* const* d_in, const int* in_sizes, int n_in,
                              void* d_out, int out_size, void* d_ws, size_t ws_size,
                              hipStream_t stream) {
    (void)in_sizes; (void)n_in; (void)out_size; (void)ws_size;
    const float* hidden = (const float*)d_in[0];
    const float* W      = (const float*)d_in[1];
    const float* bias   = (const float*)d_in[2];

    float* Bpack  = (float*)d_ws;                    // CHUNKS*32 float2 = 65536 floats
    float* logits = (float*)d_ws + CHUNKS * 64;      // NTOK*NPAD floats

    pack_B_kernel<<<(CHUNKS * 32 + 255) / 256, 256, 0, stream>>>(W, (float2*)Bpack);
    gemm_wmma_kernel<<<NTOK / BM, 256, 0, stream>>>(hidden, Bpack, bias, logits);
    route_topk_kernel<<<NTOK / 128, 128, 0, stream>>>(
        logits, (int*)d_out, (float*)d_out + (size_t)NTOK * KTOP);
}